// Loss_1022202216980
// MI455X (gfx1250) — compile-verified
//
#include <hip/hip_runtime.h>
#include <math.h>

// Problem constants (match reference).
#define HH 1024
#define WW 2048
#define TT 32
#define NPIX (HH * WW)

// Workspace layout (in floats). Total = 512 + 4096 + 2*32768 = 70144 floats (~275 KB).
#define WS_GMIN 0                    // 256 global-min partials
#define WS_GMAX 256                  // 256 global-max partials
#define WS_BMIN 512                  // 32 boxes * 64 block partials (min)
#define WS_BMAX (512 + 2048)         // 32 * 64 (max)
#define WS_P    (512 + 4096)         // P[32][1024]  per-box per-row masked sums
#define WS_PSQ  (512 + 4096 + 32768) // PSQ[32][1024] per-box per-row masked sum-of-squares

typedef float v2f __attribute__((ext_vector_type(2)));
typedef float v8f __attribute__((ext_vector_type(8)));

// ---------------------------------------------------------------------------
// Pass 1: global min/max partials. 256 blocks x 256 threads, grid-stride.
// ---------------------------------------------------------------------------
__global__ void k_gminmax(const float* __restrict__ d, float* __restrict__ ws) {
    __shared__ float smin[256], smax[256];
    const int tid = threadIdx.x;
    float mn = 3.4e38f, mx = -3.4e38f;
    for (int i = blockIdx.x * blockDim.x + tid; i < NPIX; i += gridDim.x * blockDim.x) {
        float v = d[i];
        mn = fminf(mn, v);
        mx = fmaxf(mx, v);
    }
    smin[tid] = mn; smax[tid] = mx;
    __syncthreads();
    for (int s = 128; s > 0; s >>= 1) {
        if (tid < s) {
            smin[tid] = fminf(smin[tid], smin[tid + s]);
            smax[tid] = fmaxf(smax[tid], smax[tid + s]);
        }
        __syncthreads();
    }
    if (tid == 0) {
        ws[WS_GMIN + blockIdx.x] = smin[0];
        ws[WS_GMAX + blockIdx.x] = smax[0];
    }
}

// ---------------------------------------------------------------------------
// Pass 2: per-box min/max partials. grid = (64 chunks, 32 boxes), 256 threads.
// Deterministic: every block writes its partial slot (identity if empty).
// ---------------------------------------------------------------------------
__global__ void k_boxminmax(const float* __restrict__ d, const int* __restrict__ bb,
                            float* __restrict__ ws) {
    __shared__ float smin[256], smax[256];
    const int t = blockIdx.y;
    const int x1 = bb[4 * t + 0], y1 = bb[4 * t + 1];
    const int x2 = bb[4 * t + 2], y2 = bb[4 * t + 3];
    const int bw = x2 - x1;
    const int area = bw * (y2 - y1);
    const int tid = threadIdx.x;
    float mn = 3.4e38f, mx = -3.4e38f;
    for (int i = blockIdx.x * 256 + tid; i < area; i += gridDim.x * 256) {
        int yy = i / bw, xx = i - yy * bw;
        float v = d[(size_t)(y1 + yy) * WW + (x1 + xx)];
        mn = fminf(mn, v);
        mx = fmaxf(mx, v);
    }
    smin[tid] = mn; smax[tid] = mx;
    __syncthreads();
    for (int s = 128; s > 0; s >>= 1) {
        if (tid < s) {
            smin[tid] = fminf(smin[tid], smin[tid + s]);
            smax[tid] = fmaxf(smax[tid], smax[tid + s]);
        }
        __syncthreads();
    }
    if (tid == 0) {
        ws[WS_BMIN + t * 64 + blockIdx.x] = smin[0];
        ws[WS_BMAX + t * 64 + blockIdx.x] = smax[0];
    }
}

// ---------------------------------------------------------------------------
// Pass 3 (WMMA): P[t][y] = sum_x colmask[t][x] * depth[y][x]  (and PSQ with depth^2)
// via V_WMMA_F32_16X16X4_F32:  D(16x16) = A(16x4) x B(4x16) + C.
//   A: M = box index (colmask values 0/1),  B: K = x, N = image row y.
// A layout (f32 16x4): lanes 0-15 -> M=lane, VGPR0=K0 / VGPR1=K1;
//                      lanes 16-31 -> M=lane-16, VGPR0=K2 / VGPR1=K3.
// B layout (f32 4x16): row-striped: VGPR0 = rows K0 (lanes 0-15) / K2 (lanes 16-31),
//                      VGPR1 = rows K1 / K3; N = lane & 15.
// C/D layout: VGPR r -> row M=r (lanes 0-15, N=lane) and M=r+8 (lanes 16-31).
// Per lane the two B elements are ALWAYS adjacent columns (lo: {k,k+1},
// hi: {k+2,k+3}), so one 8-byte vector load feeds both B VGPRs.
// One wave per block; no EXEC divergence (lane-dependent math via selects only).
// grid = (H/16 = 64 row-tiles, T/16 = 2 box-tiles), block = 32 (one wave32).
// ---------------------------------------------------------------------------
__global__ void k_boxgemm(const float* __restrict__ d, const int* __restrict__ bb,
                          float* __restrict__ ws) {
    const int lane  = threadIdx.x;        // 0..31
    const int m     = lane & 15;
    const int hi    = (lane >= 16) ? 1 : 0;
    const int ybase = blockIdx.x * 16;
    const int tbase = blockIdx.y * 16;

    const int t  = tbase + m;             // A-matrix row (box id) for this lane
    const int x1 = bb[4 * t + 0];
    const int x2 = bb[4 * t + 2];

    const int y = ybase + m;              // B-matrix column (image row) for this lane
    const float* __restrict__ row = d + (size_t)y * WW;

    v8f cs = {0.f, 0.f, 0.f, 0.f, 0.f, 0.f, 0.f, 0.f};
    v8f cq = {0.f, 0.f, 0.f, 0.f, 0.f, 0.f, 0.f, 0.f};

    const int koff = 2 * hi;              // this lane's first K within the 4-wide step

#pragma unroll 8
    for (int k = 0; k < WW; k += 4) {
        const int ka = k + koff;          // lane's two K values are ka, ka+1
        // Software prefetch ~2KB ahead in this lane's row (L2-resident -> warms L0).
        __builtin_prefetch(row + ka + 512, 0, 1);
        v2f a;
        a.x = (ka     >= x1 && ka     < x2) ? 1.0f : 0.0f;
        a.y = (ka + 1 >= x1 && ka + 1 < x2) ? 1.0f : 0.0f;
        const v2f b  = *(const v2f*)(row + ka);   // 8B aligned: row is 8KB-aligned, ka even
        const v2f bq = b * b;
        cs = __builtin_amdgcn_wmma_f32_16x16x4_f32(false, a, false, b,  (short)0, cs, false, false);
        cq = __builtin_amdgcn_wmma_f32_16x16x4_f32(false, a, false, bq, (short)0, cq, false, false);
    }

    // Scatter D tiles: VGPR r holds (M = r + 8*hi, N = lane&15).
    float* __restrict__ P = ws + WS_P;
    float* __restrict__ Q = ws + WS_PSQ;
    const int rowbase = hi ? 8 : 0;
#pragma unroll
    for (int r = 0; r < 8; ++r) {
        const int tt = tbase + rowbase + r;
        P[tt * HH + ybase + m] = cs[r];
        Q[tt * HH + ybase + m] = cq[r];
    }
}

// ---------------------------------------------------------------------------
// Pass 4: single block. Reduce global min/max partials, fold per-box stats
// (double accumulation over <=512 P rows), pairwise ranking loss, write out.
// ---------------------------------------------------------------------------
__global__ void k_final(const int* __restrict__ bb, const float* __restrict__ ws,
                        float* __restrict__ out) {
    __shared__ float smin[256], smax[256];
    __shared__ float us_s[TT], term_s[TT], pair_s[TT];
    const int tid = threadIdx.x;

    smin[tid] = ws[WS_GMIN + tid];
    smax[tid] = ws[WS_GMAX + tid];
    __syncthreads();
    for (int s = 128; s > 0; s >>= 1) {
        if (tid < s) {
            smin[tid] = fminf(smin[tid], smin[tid + s]);
            smax[tid] = fmaxf(smax[tid], smax[tid + s]);
        }
        __syncthreads();
    }
    const float dmin  = smin[0];
    const float range = smax[0] - dmin;

    if (tid < TT) {
        const int t  = tid;
        const int x1 = bb[4 * t + 0], y1 = bb[4 * t + 1];
        const int x2 = bb[4 * t + 2], y2 = bb[4 * t + 3];
        const double cnt = (double)(x2 - x1) * (double)(y2 - y1);

        float bmn = 3.4e38f, bmx = -3.4e38f;
        for (int i = 0; i < 64; ++i) {
            bmn = fminf(bmn, ws[WS_BMIN + t * 64 + i]);
            bmx = fmaxf(bmx, ws[WS_BMAX + t * 64 + i]);
        }

        double S = 0.0, SQ = 0.0;
        for (int y = y1; y < y2; ++y) {
            S  += (double)ws[WS_P   + t * HH + y];
            SQ += (double)ws[WS_PSQ + t * HH + y];
        }
        const double mean_raw = S / cnt;                          // raw-depth box mean
        double var_raw = (SQ - cnt * mean_raw * mean_raw) / (cnt - 1.0);
        if (var_raw < 0.0) var_raw = 0.0;

        // Affine normalization: us = (mean_raw - dmin)/range.
        us_s[t] = (float)((mean_raw - (double)dmin) / (double)range);
        // std term is normalization-invariant: std(raw)/(bmax_raw - bmin_raw).
        term_s[t] = (float)(sqrt(var_raw) / ((double)bmx - (double)bmn));
    }
    __syncthreads();

    if (tid < TT) {
        const int i = tid;
        const float invt = 1.0f / (float)TT;
        float acc = 0.0f;
        for (int j = i + 1; j < TT; ++j) {
            float val = us_s[i] - us_s[j] + (float)(j - i) * invt;
            if (val >= 0.0f) acc += val;
        }
        pair_s[i] = acc;
    }
    __syncthreads();

    if (tid == 0) {
        float la = 0.0f, ls = 0.0f;
        for (int i = 0; i < TT; ++i) { la += pair_s[i]; ls += term_s[i]; }
        out[0] = la;             // loss_acc
        out[1] = ls;             // loss_std
        out[2] = la * 1.0f + ls; // RATIO = 1.0
    }
}

// ---------------------------------------------------------------------------
extern "C" void kernel_launch(void* const* d_in, const int* in_sizes, int n_in,
                              void* d_out, int out_size, void* d_ws, size_t ws_size,
                              hipStream_t stream) {
    const float* depth = (const float*)d_in[0];   // [1,1,1024,2048] fp32
    const int*   bb    = (const int*)d_in[1];     // [32,4] int32
    float*       ws    = (float*)d_ws;            // needs >= 70144 floats (~275 KB)
    float*       out   = (float*)d_out;           // 3 floats

    k_gminmax  <<<256, 256, 0, stream>>>(depth, ws);
    k_boxminmax<<<dim3(64, TT), 256, 0, stream>>>(depth, bb, ws);
    k_boxgemm  <<<dim3(HH / 16, TT / 16), 32, 0, stream>>>(depth, bb, ws);
    k_final    <<<1, 256, 0, stream>>>(bb, ws, out);
}